// AttnDecoderRNN2_60060822667556
// MI455X (gfx1250) — compile-verified
//
#include <hip/hip_runtime.h>
#include <math.h>

#define HID 1024
#define SEQ 4096
#define VOC 50257

typedef float v2f __attribute__((ext_vector_type(2)));
typedef float v8f __attribute__((ext_vector_type(8)));

// ---------------- workspace layout (floats) ----------------
#define WS_CAT1       ((size_t)0)        // 2048    [embedded | h0]
#define WS_ATTN_PART  ((size_t)2048)     // 32*4096 partials
#define WS_ATTN       ((size_t)133120)   // 4096    pre-softmax attn logits
#define WS_APP_PART   ((size_t)137216)   // 64*1024 partials
#define WS_CAT2       ((size_t)202752)   // 2048    [embedded | attn_applied]
#define WS_COMB_PART  ((size_t)204800)   // 32*1024 partials
#define WS_X          ((size_t)237568)   // 1024    GRU input x
#define WS_H          ((size_t)238592)   // 1024    hidden state
#define WS_GI         ((size_t)239616)   // 3072
#define WS_GH         ((size_t)242688)   // 3072
#define WS_RED        ((size_t)245760)   // 2       [max, log(sumexp)]

// ---------------- reductions (wave32) ----------------
__device__ __forceinline__ float warp_sum(float v) {
#pragma unroll
  for (int off = 16; off > 0; off >>= 1) v += __shfl_down(v, off, 32);
  return v;
}
__device__ __forceinline__ float warp_max(float v) {
#pragma unroll
  for (int off = 16; off > 0; off >>= 1) v = fmaxf(v, __shfl_down(v, off, 32));
  return v;
}
__device__ float block_red(float v, bool ismax) {
  __shared__ float sred[32];
  const int lane = threadIdx.x & 31;
  const int w = threadIdx.x >> 5;
  v = ismax ? warp_max(v) : warp_sum(v);
  if (lane == 0) sred[w] = v;
  __syncthreads();
  const int nw = blockDim.x >> 5;
  if (w == 0) {
    float x = (lane < nw) ? sred[lane] : (ismax ? -INFINITY : 0.0f);
    x = ismax ? warp_max(x) : warp_sum(x);
    if (lane == 0) sred[0] = x;
  }
  __syncthreads();
  float r = sred[0];
  __syncthreads();
  return r;
}

// ---------------- kernels ----------------
__global__ void prep_kernel(const int* __restrict__ input, const float* __restrict__ hidden,
                            const float* __restrict__ emb, float* __restrict__ cat1,
                            float* __restrict__ cat2, float* __restrict__ h) {
  const int t = blockIdx.x * 256 + threadIdx.x;        // grid=4 -> t in [0,1024)
  const int idx = input[0];
  const float e = emb[(size_t)idx * HID + t];
  const float hv = hidden[t];
  cat1[t] = e;
  cat1[HID + t] = hv;
  cat2[t] = e;
  h[t] = hv;
}

// column-major GEMV partial: part[c][n] = sum_{k in chunk c of 64} x[k] * W[k*N + n]
// grid = (N/1024, K/64), block = 256; each thread owns 4 consecutive columns (b128 loads)
__global__ void gemv_col_part(const float* __restrict__ W, const float* __restrict__ x,
                              float* __restrict__ part, int N) {
  __shared__ float xs[64];
  if (threadIdx.x < 64) xs[threadIdx.x] = x[blockIdx.y * 64 + threadIdx.x];
  __syncthreads();
  const int n4 = blockIdx.x * 1024 + threadIdx.x * 4;
  const int k0 = blockIdx.y * 64;
  const float4* Wp = (const float4*)(W + (size_t)k0 * N + n4);
  const size_t rs = (size_t)(N >> 2);                  // row stride in float4
  float4 acc = make_float4(0.f, 0.f, 0.f, 0.f);
#pragma unroll 8
  for (int k = 0; k < 64; ++k) {
    const float4 w4 = Wp[(size_t)k * rs];
    const float xk = xs[k];
    acc.x = fmaf(xk, w4.x, acc.x);
    acc.y = fmaf(xk, w4.y, acc.y);
    acc.z = fmaf(xk, w4.z, acc.z);
    acc.w = fmaf(xk, w4.w, acc.w);
  }
  *(float4*)(part + (size_t)blockIdx.y * N + n4) = acc;
}

// out[n] = sum_c part[c][n] + bias[n], optional relu. grid = N/1024, block = 256 (float4)
__global__ void reduce_part(const float* __restrict__ part, const float* __restrict__ bias,
                            float* __restrict__ out, int N, int C, int do_relu) {
  const int n4 = (blockIdx.x * 256 + threadIdx.x) * 4;
  float4 acc = bias ? *(const float4*)(bias + n4) : make_float4(0.f, 0.f, 0.f, 0.f);
  for (int c = 0; c < C; ++c) {
    const float4 p = *(const float4*)(part + (size_t)c * N + n4);
    acc.x += p.x; acc.y += p.y; acc.z += p.z; acc.w += p.w;
  }
  if (do_relu) {
    acc.x = fmaxf(acc.x, 0.f); acc.y = fmaxf(acc.y, 0.f);
    acc.z = fmaxf(acc.z, 0.f); acc.w = fmaxf(acc.w, 0.f);
  }
  *(float4*)(out + n4) = acc;
}

// softmax over SEQ=4096, one block of 1024 threads
__global__ void softmax_attn(const float* __restrict__ in, float* __restrict__ out) {
  const int t = threadIdx.x;
  float vmax = -INFINITY;
  for (int i = t; i < SEQ; i += 1024) vmax = fmaxf(vmax, in[i]);
  vmax = block_red(vmax, true);
  float s = 0.0f;
  for (int i = t; i < SEQ; i += 1024) s += expf(in[i] - vmax);
  s = block_red(s, false);
  const float inv = 1.0f / s;
  for (int i = t; i < SEQ; i += 1024) out[i] = expf(in[i] - vmax) * inv;
}

// row-major GEMV (GRU): out[r] = dot(W[r, 0:1024], x) + bias[r]; 1 wave per row
__global__ void gemv_row(const float* __restrict__ W, const float* __restrict__ x,
                         const float* __restrict__ bias, float* __restrict__ out, int R) {
  __shared__ float xs[HID];
  for (int i = threadIdx.x; i < HID; i += 256) xs[i] = x[i];
  __syncthreads();
  const int lane = threadIdx.x & 31;
  const int wave = threadIdx.x >> 5;
  const int row = blockIdx.x * 8 + wave;
  if (row >= R) return;
  const float4* Wr = (const float4*)(W + (size_t)row * HID);
  const float4* xv = (const float4*)xs;
  float acc = 0.0f;
#pragma unroll
  for (int i = lane; i < HID / 4; i += 32) {
    const float4 w4 = Wr[i];
    const float4 x4 = xv[i];
    acc += w4.x * x4.x + w4.y * x4.y + w4.z * x4.z + w4.w * x4.w;
  }
  acc = warp_sum(acc);
  if (lane == 0) out[row] = acc + bias[row];
}

// GRU gate math; updates h in place, x = relu(h'), mirrors h' to final output slot
__global__ void gru_gate(const float* __restrict__ gi, const float* __restrict__ gh,
                         float* __restrict__ h, float* __restrict__ x,
                         float* __restrict__ h_out) {
  const int t = blockIdx.x * 256 + threadIdx.x;        // grid=4 -> t in [0,1024)
  const float ir = gi[t], iz = gi[HID + t], in_ = gi[2 * HID + t];
  const float hr = gh[t], hz = gh[HID + t], hn = gh[2 * HID + t];
  const float r = 1.0f / (1.0f + expf(-(ir + hr)));
  const float z = 1.0f / (1.0f + expf(-(iz + hz)));
  const float n = tanhf(in_ + r * hn);
  const float hnew = (1.0f - z) * n + z * h[t];
  h[t] = hnew;
  x[t] = fmaxf(hnew, 0.0f);
  h_out[t] = hnew;
}

// Output projection: logits[v] = dot(h, out_W[:, v]) + out_b[v], out_W row-major [1024, VOC].
// WMMA f32 16x16x4: one wave per 16-column tile; h in A (all lanes load unconditionally --
// A rows 1..15 only feed D rows we never read, so no masking / exec churn needed).
// K processed in 32-row batches: 8 ds_b64 + 16 global_b32 issued as clauses, 1 wait, 8 WMMAs.
__global__ void out_logits_kernel(const float* __restrict__ W, const float* __restrict__ h,
                                  const float* __restrict__ bias, float* __restrict__ out) {
  __shared__ float hs[HID];
  for (int i = threadIdx.x; i < HID; i += 256) hs[i] = h[i];
  __syncthreads();
  const int lane = threadIdx.x & 31;
  const int wave = threadIdx.x >> 5;
#if __has_builtin(__builtin_amdgcn_wmma_f32_16x16x4_f32)
  const int ntiles = (VOC + 15) / 16;
  const int tile = blockIdx.x * 8 + wave;
  if (tile >= ntiles) return;                 // uniform whole-wave exit; EXEC all-1 at WMMA
  const int col = tile * 16 + (lane & 15);
  const int colc = (col < VOC) ? col : (VOC - 1);   // clamp addr; extra lanes never stored
  const int khi = (lane >> 4) << 1;                 // 0 for lanes 0-15, 2 for lanes 16-31
  const float2* h2 = (const float2*)hs;
  const int hofs = khi >> 1;                        // float2 index offset (0 or 1)
  const float* base = W + (size_t)khi * VOC + colc;
  v8f acc = {};
  for (int k = 0; k < HID; k += 32) {
    const float* pk = base + (size_t)k * VOC;
    __builtin_prefetch(pk + (size_t)32 * VOC, 0, 1);   // speculative; OOB tail dropped
    float2 A[8];
    float bx[8], by[8];
#pragma unroll
    for (int j = 0; j < 8; ++j) {
      A[j] = h2[(k >> 1) + hofs + 2 * j];             // {h[kk], h[kk+1]} : ds_load_b64
      const float* pj = pk + (size_t)(4 * j) * VOC;
      bx[j] = pj[0];                                  // B[kk,   n]
      by[j] = pj[VOC];                                // B[kk+1, n]
    }
#pragma unroll
    for (int j = 0; j < 8; ++j) {
      v2f a, b;
      a.x = A[j].x; a.y = A[j].y;
      b.x = bx[j];  b.y = by[j];
      acc = __builtin_amdgcn_wmma_f32_16x16x4_f32(false, a, false, b, (short)0, acc,
                                                  false, false);
    }
  }
  // D row M=0 = VGPR0, lanes 0-15
  if (lane < 16 && col < VOC) out[col] = acc[0] + bias[col];
#else
  (void)lane; (void)wave;
  const int v = blockIdx.x * 128 + threadIdx.x;     // fallback: 128 cols per block
  if (threadIdx.x < 128 && v < VOC) {
    float acc = bias[v];
#pragma unroll 8
    for (int k = 0; k < HID; ++k) acc = fmaf(hs[k], W[(size_t)k * VOC + v], acc);
    out[v] = acc;
  }
#endif
}

// single-block max + log-sum-exp over VOC logits
__global__ void lse_kernel(const float* __restrict__ logits, float* __restrict__ red) {
  const int t = threadIdx.x;                         // 1024 threads
  float vmax = -INFINITY;
  for (int i = t; i < VOC; i += 1024) vmax = fmaxf(vmax, logits[i]);
  vmax = block_red(vmax, true);
  float s = 0.0f;
  for (int i = t; i < VOC; i += 1024) s += expf(logits[i] - vmax);
  s = block_red(s, false);
  if (t == 0) { red[0] = vmax; red[1] = logf(s); }
}

__global__ void logsoftmax_apply(float* __restrict__ logits, const float* __restrict__ red) {
  const int v = blockIdx.x * 256 + threadIdx.x;
  if (v < VOC) logits[v] -= (red[0] + red[1]);
}

// ---------------- host launcher ----------------
extern "C" void kernel_launch(void* const* d_in, const int* in_sizes, int n_in,
                              void* d_out, int out_size, void* d_ws, size_t ws_size,
                              hipStream_t stream) {
  (void)in_sizes; (void)n_in; (void)out_size; (void)ws_size;
  const int*   input  = (const int*)  d_in[0];
  const float* hidden = (const float*)d_in[1];
  const float* enc    = (const float*)d_in[2];
  const float* emb    = (const float*)d_in[3];
  const float* attn_W = (const float*)d_in[4];
  const float* attn_b = (const float*)d_in[5];
  const float* comb_W = (const float*)d_in[6];
  const float* comb_b = (const float*)d_in[7];
  const float* gWih   = (const float*)d_in[8];
  const float* gWhh   = (const float*)d_in[9];
  const float* gbih   = (const float*)d_in[10];
  const float* gbhh   = (const float*)d_in[11];
  const float* out_W  = (const float*)d_in[12];
  const float* out_b  = (const float*)d_in[13];

  float* out = (float*)d_out;               // [log_probs VOC | h HID | attn SEQ]
  float* ws  = (float*)d_ws;
  float* cat1  = ws + WS_CAT1;
  float* attnP = ws + WS_ATTN_PART;
  float* attnL = ws + WS_ATTN;
  float* appP  = ws + WS_APP_PART;
  float* cat2  = ws + WS_CAT2;
  float* combP = ws + WS_COMB_PART;
  float* xbuf  = ws + WS_X;
  float* hbuf  = ws + WS_H;
  float* gibuf = ws + WS_GI;
  float* ghbuf = ws + WS_GH;
  float* red   = ws + WS_RED;
  float* attn_out = out + VOC + HID;

  prep_kernel<<<4, 256, 0, stream>>>(input, hidden, emb, cat1, cat2, hbuf);

  // attn logits: W [2048 x 4096], K chunks of 64 -> grid (4, 32)
  gemv_col_part<<<dim3(SEQ / 1024, 2048 / 64), 256, 0, stream>>>(attn_W, cat1, attnP, SEQ);
  reduce_part<<<SEQ / 1024, 256, 0, stream>>>(attnP, attn_b, attnL, SEQ, 32, 0);
  softmax_attn<<<1, 1024, 0, stream>>>(attnL, attn_out);

  // attn_applied: enc [4096 x 1024] -> grid (1, 64)
  gemv_col_part<<<dim3(HID / 1024, SEQ / 64), 256, 0, stream>>>(enc, attn_out, appP, HID);
  reduce_part<<<HID / 1024, 256, 0, stream>>>(appP, nullptr, cat2 + HID, HID, 64, 0);

  // combine: [2048 x 1024] -> grid (1, 32); relu fused -> x
  gemv_col_part<<<dim3(HID / 1024, 2048 / 64), 256, 0, stream>>>(comb_W, cat2, combP, HID);
  reduce_part<<<HID / 1024, 256, 0, stream>>>(combP, comb_b, xbuf, HID, 32, 1);

  // 2 GRU layers (sequential dependency)
  for (int l = 0; l < 2; ++l) {
    gemv_row<<<(3 * HID) / 8, 256, 0, stream>>>(gWih, xbuf, gbih, gibuf, 3 * HID);
    gemv_row<<<(3 * HID) / 8, 256, 0, stream>>>(gWhh, hbuf, gbhh, ghbuf, 3 * HID);
    gru_gate<<<HID / 256, 256, 0, stream>>>(gibuf, ghbuf, hbuf, xbuf, out + VOC);
  }

  // output projection (WMMA) + log_softmax
  const int ntiles = (VOC + 15) / 16;
  out_logits_kernel<<<(ntiles + 7) / 8, 256, 0, stream>>>(out_W, hbuf, out_b, out);
  lse_kernel<<<1, 1024, 0, stream>>>(out, red);
  logsoftmax_apply<<<(VOC + 255) / 256, 256, 0, stream>>>(out, red);
}